// FixedPositionalEncoding_2d_17437567222345
// MI455X (gfx1250) — compile-verified
//
#include <hip/hip_runtime.h>
#include <hip/hip_bf16.h>

// Native vector type so __builtin_nontemporal_load/store accept it.
typedef float floatx4 __attribute__((ext_vector_type(4)));

#define D_MODEL   512
#define HW        10000          // 100*100
#define D4        (D_MODEL / 4)  // 128 float4 per patch row

// ---------------------------------------------------------------------------
// Kernel 1: LDS-tiled transpose  pe[d, hw] (512 x 10000)  ->  pe_t[hw, d]
// Both sides coalesced; 32x33 tile padding kills LDS bank conflicts.
// Cost ~41 MB of traffic (~1.8 us) — noise vs the 22 us main pass.
// ---------------------------------------------------------------------------
__global__ void pe_transpose_kernel(const float* __restrict__ pe,
                                    float* __restrict__ pe_t) {
    __shared__ float tile[32][33];
    const int d0  = blockIdx.x * 32;   // gridDim.x = 16  -> d  in [0,512)
    const int hw0 = blockIdx.y * 32;   // gridDim.y = 313 -> hw in [0,10016)
    const int tx = threadIdx.x;
    const int ty = threadIdx.y;

    const int hw_r = hw0 + tx;         // coalesced read along hw
    const int d_r  = d0 + ty;
    if (hw_r < HW)
        tile[ty][tx] = pe[d_r * HW + hw_r];

    __syncthreads();

    const int hw_w = hw0 + ty;
    const int d_w  = d0 + tx;          // coalesced write along d
    if (hw_w < HW)
        pe_t[hw_w * D_MODEL + d_w] = tile[tx][ty];
}

// ---------------------------------------------------------------------------
// Kernel 2: main pass. One float4 per thread. Waves are fully contained in a
// single patch (128 float4-threads per patch, wave32), so coord loads are
// wave-uniform. x/out are streamed with non-temporal hints (th:TH_NT) so the
// 20 MB pe_t stays resident in the 192 MB L2; pe_t loads are regular-temporal
// coalesced b128 that hit L2.
// ---------------------------------------------------------------------------
__global__ void pe_add_kernel(const floatx4* __restrict__ x,
                              const float*   __restrict__ coord,
                              const floatx4* __restrict__ pe_t,
                              floatx4*       __restrict__ out,
                              int total4) {
    const int tid = blockIdx.x * blockDim.x + threadIdx.x;
    if (tid >= total4) return;

    const int patch = tid >> 7;        // / 128
    const int d4    = tid & 127;

    // coord values are integral floats in [0, 10000); reference does
    // (coord / 100.0).astype(int32) -> IEEE divide then truncate.
    const float cH = coord[2 * patch];
    const float cW = coord[2 * patch + 1];
    const int i0 = (int)(cH / 100.0f);
    const int i1 = (int)(cW / 100.0f);
    const int cell = i0 * 100 + i1;

    // Streaming operand: non-temporal load (don't pollute L2).
    const floatx4 xv = __builtin_nontemporal_load(x + tid);
    // Gathered operand: regular-temporal, coalesced, L2-resident.
    const floatx4 pv = pe_t[cell * D4 + d4];

    const floatx4 ov = xv + 0.1f * pv;

    // Streaming result: non-temporal store.
    __builtin_nontemporal_store(ov, out + tid);
}

// ---------------------------------------------------------------------------
// Fallback (workspace too small): direct strided gather from pe. Correct but
// request-rate-bound on the gather.
// ---------------------------------------------------------------------------
__global__ void pe_add_fallback_kernel(const floatx4* __restrict__ x,
                                       const float*   __restrict__ coord,
                                       const float*   __restrict__ pe,
                                       floatx4*       __restrict__ out,
                                       int total4) {
    const int tid = blockIdx.x * blockDim.x + threadIdx.x;
    if (tid >= total4) return;

    const int patch = tid >> 7;
    const int d4    = tid & 127;

    const float cH = coord[2 * patch];
    const float cW = coord[2 * patch + 1];
    const int i0 = (int)(cH / 100.0f);
    const int i1 = (int)(cW / 100.0f);
    const int cell = i0 * 100 + i1;
    const int dbase = d4 * 4;

    floatx4 pv;
    pv.x = pe[(dbase + 0) * HW + cell];
    pv.y = pe[(dbase + 1) * HW + cell];
    pv.z = pe[(dbase + 2) * HW + cell];
    pv.w = pe[(dbase + 3) * HW + cell];

    const floatx4 xv = __builtin_nontemporal_load(x + tid);
    const floatx4 ov = xv + 0.1f * pv;
    __builtin_nontemporal_store(ov, out + tid);
}

extern "C" void kernel_launch(void* const* d_in, const int* in_sizes, int n_in,
                              void* d_out, int out_size, void* d_ws, size_t ws_size,
                              hipStream_t stream) {
    const float* x     = (const float*)d_in[0];   // (16, 8192, 512) fp32
    const float* coord = (const float*)d_in[1];   // (16, 8192, 2)  fp32
    const float* pe    = (const float*)d_in[2];   // (512, 100, 100) fp32
    float* out = (float*)d_out;                   // (16, 8192, 512) fp32

    const int total4 = out_size / 4;              // 16,777,216 float4
    const int threads = 256;                      // 8 waves on wave32
    const int blocks = (total4 + threads - 1) / threads;

    const size_t pe_t_bytes = (size_t)HW * D_MODEL * sizeof(float); // 20.48 MB

    if (ws_size >= pe_t_bytes) {
        float* pe_t = (float*)d_ws;
        dim3 tb(32, 32);
        dim3 tg(D_MODEL / 32, (HW + 31) / 32);    // 16 x 313
        pe_transpose_kernel<<<tg, tb, 0, stream>>>(pe, pe_t);
        pe_add_kernel<<<blocks, threads, 0, stream>>>(
            (const floatx4*)x, coord, (const floatx4*)pe_t,
            (floatx4*)out, total4);
    } else {
        pe_add_fallback_kernel<<<blocks, threads, 0, stream>>>(
            (const floatx4*)x, coord, pe, (floatx4*)out, total4);
    }
}